// Hausdorff_2044404433132
// MI455X (gfx1250) — compile-verified
//
#include <hip/hip_runtime.h>
#include <hip/hip_bf16.h>

typedef float v2f __attribute__((ext_vector_type(2)));
typedef float v8f __attribute__((ext_vector_type(8)));

#define BATCH 8
#define NPTS  4096
#define ROWS_PER_WG 128                      // 8 waves x 16 rows
#define WGS_PER_BATCH (NPTS / ROWS_PER_WG)   // 32
#define CSTEPS (NPTS / 32)                   // 128 iterations of 2 column tiles
#define POS_INF_BITS 0x7F800000u

// Raw CDNA5 min/max — skips clang's canonicalize of WMMA outputs (inputs are
// finite by construction, NaN handling is irrelevant here).
__device__ __forceinline__ float min3f(float a, float b, float c) {
    float d;
    asm("v_min3_num_f32 %0, %1, %2, %3" : "=v"(d) : "v"(a), "v"(b), "v"(c));
    return d;
}
__device__ __forceinline__ float min2f(float a, float b) {
    float d;
    asm("v_min_num_f32 %0, %1, %2" : "=v"(d) : "v"(a), "v"(b));
    return d;
}
__device__ __forceinline__ float max0f(float a) {
    float d;
    asm("v_max_num_f32 %0, 0, %1" : "=v"(d) : "v"(a));
    return d;
}

// ---------------------------------------------------------------------------
// Kernel 0: init workspace (colmin = +inf, h_ab = 0). Must run every call.
// ---------------------------------------------------------------------------
__global__ __launch_bounds__(256) void hd_init(unsigned* __restrict__ colmin,
                                               unsigned* __restrict__ hab) {
    int idx = blockIdx.x * 256 + threadIdx.x;
    if (idx < BATCH * NPTS) colmin[idx] = POS_INF_BITS;
    if (idx < BATCH)        hab[idx]    = 0u;
}

// ---------------------------------------------------------------------------
// Kernel 1: main pass. One workgroup = 128 pred rows of one batch; each wave
// owns 16 rows and sweeps all target columns, 2 WMMA tiles per iteration.
//
// Fused distance WMMA (V_WMMA_F32_16X16X4_F32, full K=4 used):
//   A row i  = (-2*px_i, -2*py_i, |p_i|^2, 1)   [lanes 0-15: K=0,1 | 16-31: K=2,3]
//   B col j  = ( qx_j,    qy_j,   1, |q_j|^2)
//   => C[i][j] = |p_i|^2 + |q_j|^2 - 2*(p_i.q_j) = dist2  (f32 accumulate)
//
// max(.,0) (cancellation clamp) commutes with min-reductions, so it is applied
// only to the per-tile column mins (required before the float->uint bitcast)
// and once to the final row max.
// ---------------------------------------------------------------------------
__global__ __launch_bounds__(256) void hd_main(const float2* __restrict__ pred,
                                               const float2* __restrict__ tgt,
                                               unsigned* __restrict__ colmin,
                                               unsigned* __restrict__ hab) {
    __shared__ unsigned cmLDS[NPTS];   // 16 KB: col-min over this WG's 128 rows

    const int b       = blockIdx.x / WGS_PER_BATCH;
    const int rowBase = (blockIdx.x % WGS_PER_BATCH) * ROWS_PER_WG;
    const int wave    = threadIdx.x >> 5;
    const int lane    = threadIdx.x & 31;
    const int half    = lane >> 4;      // 0: K=0,1 carrier | 1: K=2,3 carrier
    const int lane16  = lane & 15;

    for (int k = threadIdx.x; k < NPTS; k += 256) cmLDS[k] = POS_INF_BITS;
    __syncthreads();

    const float2* predb = pred + b * NPTS;
    const float2* tgtb  = tgt  + b * NPTS;
    const int i0 = rowBase + wave * 16;

    // A fragment: row = lane16 in both halves.
    float2 ap = predb[i0 + lane16];
    float pn  = ap.x * ap.x + ap.y * ap.y;
    v2f afrag;
    afrag[0] = half ? pn   : -2.0f * ap.x;
    afrag[1] = half ? 1.0f : -2.0f * ap.y;

    float rmin[8];
#pragma unroll
    for (int v = 0; v < 8; ++v) rmin[v] = 3.4e38f;

    for (int jt = 0; jt < CSTEPS; ++jt) {
        const int j0 = jt * 32;
        // two target points per lane: columns j0+2*lane16 (tile0), +1 (tile1)
        const float4 q = *(const float4*)(tgtb + j0 + 2 * lane16);
        if (jt + 1 < CSTEPS)
            __builtin_prefetch(tgtb + j0 + 32 + 2 * lane16, 0, 3);

        float qn0 = q.x * q.x + q.y * q.y;
        float qn1 = q.z * q.z + q.w * q.w;

        v2f b0, b1;
        b0[0] = half ? 1.0f : q.x;  b0[1] = half ? qn0 : q.y;
        b1[0] = half ? 1.0f : q.z;  b1[1] = half ? qn1 : q.w;

        v8f c0 = {}, c1 = {};
        c0 = __builtin_amdgcn_wmma_f32_16x16x4_f32(false, afrag, false, b0,
                                                   (short)0, c0, false, false);
        c1 = __builtin_amdgcn_wmma_f32_16x16x4_f32(false, afrag, false, b1,
                                                   (short)0, c1, false, false);

        // Row mins: one v_min3 per accumulator VGPR.
#pragma unroll
        for (int v = 0; v < 8; ++v)
            rmin[v] = min3f(rmin[v], c0[v], c1[v]);

        // Column mins: 4-op v_min3 trees, then clamp before the uint bitcast.
        float cm0 = min2f(min3f(min3f(c0[0], c0[1], c0[2]), c0[6], c0[7]),
                          min3f(c0[3], c0[4], c0[5]));
        float cm1 = min2f(min3f(min3f(c1[0], c1[1], c1[2]), c1[6], c1[7]),
                          min3f(c1[3], c1[4], c1[5]));
        cm0 = max0f(cm0);
        cm1 = max0f(cm1);

        // Both lane halves hold the same columns (different row groups):
        // the LDS atomic does the cross-half combine — no shuffle needed.
        atomicMin(&cmLDS[j0 + 2 * lane16],     __float_as_uint(cm0));
        atomicMin(&cmLDS[j0 + 2 * lane16 + 1], __float_as_uint(cm1));
    }

    // Row-min: butterfly min within each 16-lane half (lane = column j mod 16)
#pragma unroll
    for (int v = 0; v < 8; ++v) {
        float r = rmin[v];
        r = min2f(r, __shfl_xor(r, 1, 32));
        r = min2f(r, __shfl_xor(r, 2, 32));
        r = min2f(r, __shfl_xor(r, 4, 32));
        r = min2f(r, __shfl_xor(r, 8, 32));
        rmin[v] = r;
    }
    float t = fmaxf(rmin[0], rmin[1]);
#pragma unroll
    for (int v = 2; v < 8; ++v) t = fmaxf(t, rmin[v]);
    t = fmaxf(t, __shfl_xor(t, 16, 32));   // max over both halves' rows
    t = max0f(t);                          // single cancellation clamp
    if (lane == 0) atomicMax(&hab[b], __float_as_uint(t));

    __syncthreads();
    // Flush workgroup col-mins to global (16 entries/thread)
    for (int k = threadIdx.x; k < NPTS; k += 256)
        atomicMin(&colmin[b * NPTS + k], cmLDS[k]);
}

// ---------------------------------------------------------------------------
// Kernel 2: finalize. Single block: h_ba = max_j colmin, score = sqrt(max),
// mean over batches.
// ---------------------------------------------------------------------------
__global__ __launch_bounds__(256) void hd_final(const unsigned* __restrict__ colmin,
                                                const unsigned* __restrict__ hab,
                                                float* __restrict__ out) {
    __shared__ float red[256];
    float sum = 0.0f;
    for (int b = 0; b < BATCH; ++b) {
        float m = 0.0f;
        for (int k = threadIdx.x; k < NPTS; k += 256)
            m = fmaxf(m, __uint_as_float(colmin[b * NPTS + k]));
        red[threadIdx.x] = m;
        __syncthreads();
        for (int off = 128; off > 0; off >>= 1) {
            if (threadIdx.x < off)
                red[threadIdx.x] = fmaxf(red[threadIdx.x], red[threadIdx.x + off]);
            __syncthreads();
        }
        if (threadIdx.x == 0) {
            float score_sq = fmaxf(red[0], __uint_as_float(hab[b]));
            sum += sqrtf(score_sq);
        }
        __syncthreads();
    }
    if (threadIdx.x == 0) out[0] = sum * (1.0f / BATCH);
}

// ---------------------------------------------------------------------------
extern "C" void kernel_launch(void* const* d_in, const int* in_sizes, int n_in,
                              void* d_out, int out_size, void* d_ws, size_t ws_size,
                              hipStream_t stream) {
    const float2* pred = (const float2*)d_in[0];
    const float2* tgt  = (const float2*)d_in[1];
    unsigned* colmin = (unsigned*)d_ws;            // BATCH*NPTS uints = 128 KB
    unsigned* hab    = colmin + BATCH * NPTS;      // BATCH uints
    float* out = (float*)d_out;

    hd_init<<<(BATCH * NPTS + 255) / 256, 256, 0, stream>>>(colmin, hab);
    hd_main<<<BATCH * WGS_PER_BATCH, 256, 0, stream>>>(pred, tgt, colmin, hab);
    hd_final<<<1, 256, 0, stream>>>(colmin, hab, out);
}